// BTGNNs_88098369176166
// MI455X (gfx1250) — compile-verified
//
#include <hip/hip_runtime.h>

#define NN 50000        // nodes
#define NE 800000       // edges
#define DIM 256
#define HEADS 8
#define HEAD_DIM 32
#define NEG_SLOPE 0.2f

typedef float v2f __attribute__((ext_vector_type(2)));
typedef float v8f __attribute__((ext_vector_type(8)));

// ---------------------------------------------------------------------------
// GEMM: out[N,256] = X[N,256] @ W[256,256] via V_WMMA_F32_16X16X4_F32.
// Block = 256 threads = 8 waves. Each block owns one 16-row M tile; wave w
// owns column tiles [32w, 32w+16) and [32w+16, 32w+32). 64 K-steps of 4.
// A-frag layout (ISA 7.12.2, 32-bit A 16x4): lane&15 = M row, lane>>4 picks
// K pair {0,1} vs {2,3}  -> contiguous v2f load from row-major X.
// B-frag: lane&15 = N col, lane>>4 picks K row pair. C: VGPR i -> row
// row0 + 8*(lane>>4) + i, col = tile + (lane&15).
// ---------------------------------------------------------------------------
__global__ __launch_bounds__(256)
void gat_gemm(const float* __restrict__ X, const float* __restrict__ W,
              float* __restrict__ out) {
  const int wave = threadIdx.x >> 5;
  const int lane = threadIdx.x & 31;
  const int half = lane >> 4;
  const int l    = lane & 15;
  const int row0 = blockIdx.x * 16;

  const int colA = wave * 32 + l;        // first N tile column for this lane
  const int colB = colA + 16;            // second N tile column

  v8f c0 = {};
  v8f c1 = {};

  const float* xr = X + (row0 + l) * DIM;   // this lane's A row
  for (int k0 = 0; k0 < DIM; k0 += 4) {
    const int kA = k0 + half * 2;
    v2f a = *(const v2f*)(xr + kA);                 // X[row][kA], X[row][kA+1]
    const float* w0 = W + kA * DIM;
    v2f b0 = { w0[colA], w0[DIM + colA] };          // W[kA][col], W[kA+1][col]
    v2f b1 = { w0[colB], w0[DIM + colB] };
    c0 = __builtin_amdgcn_wmma_f32_16x16x4_f32(false, a, false, b0,
                                               (short)0, c0, false, false);
    c1 = __builtin_amdgcn_wmma_f32_16x16x4_f32(false, a, false, b1,
                                               (short)0, c1, false, false);
  }

  float* o = out + (row0 + half * 8) * DIM;
#pragma unroll
  for (int i = 0; i < 8; ++i) {
    o[i * DIM + colA] = c0[i];
    o[i * DIM + colB] = c1[i];
  }
}

// ---------------------------------------------------------------------------
// el[n,h] = dot(feat[n, h*32:(h+1)*32], al[h]);  er likewise.
// One wave32 per (node, head): lane = position within the head slice.
// ---------------------------------------------------------------------------
__global__ __launch_bounds__(256)
void gat_attn(const float* __restrict__ feat, const float* __restrict__ al,
              const float* __restrict__ ar, float* __restrict__ el,
              float* __restrict__ er) {
  const int gw   = (blockIdx.x * blockDim.x + threadIdx.x) >> 5;
  const int lane = threadIdx.x & 31;
  if (gw >= NN * HEADS) return;
  const int node = gw >> 3;
  const int head = gw & 7;

  const float f  = feat[node * DIM + head * HEAD_DIM + lane];
  float vl = f * al[head * HEAD_DIM + lane];
  float vr = f * ar[head * HEAD_DIM + lane];
#pragma unroll
  for (int off = 16; off > 0; off >>= 1) {
    vl += __shfl_xor(vl, off);
    vr += __shfl_xor(vr, off);
  }
  if (lane == 0) {
    el[gw] = vl;
    er[gw] = vr;
  }
}

// Zero the aggregation target and init per-(node,head) softmax stats.
__global__ __launch_bounds__(256)
void gat_init(float* __restrict__ nmax, float* __restrict__ nsum,
              float* __restrict__ agg) {
  const int t = blockIdx.x * blockDim.x + threadIdx.x;
  if (t < NN * DIM) agg[t] = 0.0f;
  if (t < NN * HEADS) {
    nmax[t] = -__builtin_inff();
    nsum[t] = 0.0f;
  }
}

// Monotonic int-punned float atomic max (target initialized to -inf bits).
__device__ inline void atomicMaxF32(float* addr, float val) {
  if (val >= 0.0f)
    atomicMax((int*)addr, __float_as_int(val));
  else
    atomicMin((unsigned int*)addr, (unsigned int)__float_as_int(val));
}

// Pass 1: e = leakyrelu(el[src] + er[dst]); store; segment-max into nmax[dst].
__global__ __launch_bounds__(256)
void gat_edge_logits(const int* __restrict__ src, const int* __restrict__ dst,
                     const float* __restrict__ el, const float* __restrict__ er,
                     float* __restrict__ ebuf, float* __restrict__ nmax) {
  const int t = blockIdx.x * blockDim.x + threadIdx.x;
  if (t >= NE * HEADS) return;
  const int e = t >> 3;
  const int h = t & 7;
  const int s = src[e];
  const int d = dst[e];
  float v = el[s * HEADS + h] + er[d * HEADS + h];
  v = v > 0.0f ? v : NEG_SLOPE * v;
  ebuf[t] = v;
  atomicMaxF32(&nmax[d * HEADS + h], v);
}

// Pass 2: a = exp(e - nmax[dst]); store in place; segment-sum into nsum[dst].
__global__ __launch_bounds__(256)
void gat_edge_exp(const int* __restrict__ dst, float* __restrict__ ebuf,
                  const float* __restrict__ nmax, float* __restrict__ nsum) {
  const int t = blockIdx.x * blockDim.x + threadIdx.x;
  if (t >= NE * HEADS) return;
  const int e = t >> 3;
  const int h = t & 7;
  const int d = dst[e];
  const float a = expf(ebuf[t] - nmax[d * HEADS + h]);
  ebuf[t] = a;
  atomicAdd(&nsum[d * HEADS + h], a);
}

// Pass 3: out[dst] += feat[src] * (a / nsum[dst]) per head.
// One wave32 per edge; lane handles 8 consecutive features (head = lane/4).
// Wave reads 1 KB contiguous of feat[src] (L2-resident), scatter via L2 atomics.
__global__ __launch_bounds__(256)
void gat_edge_aggregate(const int* __restrict__ src, const int* __restrict__ dst,
                        const float* __restrict__ ebuf,
                        const float* __restrict__ nsum,
                        const float* __restrict__ feat, float* __restrict__ out) {
  const int gw   = (blockIdx.x * blockDim.x + threadIdx.x) >> 5;
  const int lane = threadIdx.x & 31;
  if (gw >= NE) return;
  const int s = src[gw];
  const int d = dst[gw];
  const int h = lane >> 2;   // (lane*8)/32

  const float alpha = ebuf[gw * HEADS + h] / nsum[d * HEADS + h];

  const float4* fs = (const float4*)(feat + s * DIM + lane * 8);
  const float4 f0 = fs[0];
  const float4 f1 = fs[1];

  float* o = out + d * DIM + lane * 8;
  atomicAdd(o + 0, f0.x * alpha);
  atomicAdd(o + 1, f0.y * alpha);
  atomicAdd(o + 2, f0.z * alpha);
  atomicAdd(o + 3, f0.w * alpha);
  atomicAdd(o + 4, f1.x * alpha);
  atomicAdd(o + 5, f1.y * alpha);
  atomicAdd(o + 6, f1.z * alpha);
  atomicAdd(o + 7, f1.w * alpha);
}

// out = relu(acc + bias), in place.
__global__ __launch_bounds__(256)
void gat_bias_relu(float* __restrict__ acc, const float* __restrict__ b) {
  const int t = blockIdx.x * blockDim.x + threadIdx.x;
  if (t >= NN * DIM) return;
  const float v = acc[t] + b[t & (DIM - 1)];
  acc[t] = v > 0.0f ? v : 0.0f;
}

extern "C" void kernel_launch(void* const* d_in, const int* in_sizes, int n_in,
                              void* d_out, int out_size, void* d_ws, size_t ws_size,
                              hipStream_t stream) {
  const float* data = (const float*)d_in[0];
  const int*   src  = (const int*)d_in[1];
  const int*   dst  = (const int*)d_in[2];
  const float* W1   = (const float*)d_in[3];
  const float* al1  = (const float*)d_in[4];
  const float* ar1  = (const float*)d_in[5];
  const float* b1   = (const float*)d_in[6];
  const float* W2   = (const float*)d_in[7];
  const float* al2  = (const float*)d_in[8];
  const float* ar2  = (const float*)d_in[9];
  const float* b2   = (const float*)d_in[10];
  float* out = (float*)d_out;

  float* p = (float*)d_ws;
  float* feat = p;  p += (size_t)NN * DIM;       // 51.2 MB
  float* h    = p;  p += (size_t)NN * DIM;       // 51.2 MB
  float* el   = p;  p += (size_t)NN * HEADS;
  float* er   = p;  p += (size_t)NN * HEADS;
  float* nmax = p;  p += (size_t)NN * HEADS;
  float* nsum = p;  p += (size_t)NN * HEADS;
  float* ebuf = p;  p += (size_t)NE * HEADS;     // 25.6 MB

  const int gGemm = NN / 16;                       // 3125
  const int gAttn = (NN * HEADS * 32) / 256;       // 50000
  const int gNode = (NN * DIM) / 256;              // 50000
  const int gEdgeH = (NE * HEADS) / 256;           // 25000
  const int gEdgeW = (NE * 32) / 256;              // 100000

  auto layer = [&](const float* xin, const float* W, const float* al,
                   const float* ar, const float* bias, float* agg) {
    gat_gemm<<<gGemm, 256, 0, stream>>>(xin, W, feat);
    gat_attn<<<gAttn, 256, 0, stream>>>(feat, al, ar, el, er);
    gat_init<<<gNode, 256, 0, stream>>>(nmax, nsum, agg);
    gat_edge_logits<<<gEdgeH, 256, 0, stream>>>(src, dst, el, er, ebuf, nmax);
    gat_edge_exp<<<gEdgeH, 256, 0, stream>>>(dst, ebuf, nmax, nsum);
    gat_edge_aggregate<<<gEdgeW, 256, 0, stream>>>(src, dst, ebuf, nsum, feat, agg);
    gat_bias_relu<<<gNode, 256, 0, stream>>>(agg, bias);
  };

  layer(data, W1, al1, ar1, b1, h);    // layer 1 -> h
  layer(h,    W2, al2, ar2, b2, out);  // layer 2 -> d_out
}